// PointPWC_14714557956153
// MI455X (gfx1250) — compile-verified
//
#include <hip/hip_runtime.h>

typedef __attribute__((ext_vector_type(2))) float v2f;
typedef __attribute__((ext_vector_type(8))) float v8f;

#if __has_builtin(__builtin_amdgcn_wmma_f32_16x16x4_f32)
#define USE_WMMA44 1
#endif

// ---------------------------------------------------------------------------
// prep: build pc2 = coords+gt, warp = coords+pred, and the "prepared" DB
// tables (-2x,-2y,-2z,||p||^2) so WMMA emits score = ||p||^2 - 2 q.p directly.
// ---------------------------------------------------------------------------
__global__ __launch_bounds__(256) void prep_kernel(
    const float* __restrict__ pred, const float* __restrict__ gt,
    const float* __restrict__ coords, int N,
    float* __restrict__ pc2, float* __restrict__ warp,
    float4* __restrict__ pc1T, float4* __restrict__ pc2T,
    float4* __restrict__ warpT, float* __restrict__ nwarp,
    float* __restrict__ npc2) {
  int i = blockIdx.x * blockDim.x + threadIdx.x;
  if (i >= N) return;
  float cx = coords[3 * i + 0], cy = coords[3 * i + 1], cz = coords[3 * i + 2];
  float p2x = cx + gt[3 * i + 0], p2y = cy + gt[3 * i + 1], p2z = cz + gt[3 * i + 2];
  float wx = cx + pred[3 * i + 0], wy = cy + pred[3 * i + 1], wz = cz + pred[3 * i + 2];
  pc2[3 * i + 0] = p2x; pc2[3 * i + 1] = p2y; pc2[3 * i + 2] = p2z;
  warp[3 * i + 0] = wx; warp[3 * i + 1] = wy; warp[3 * i + 2] = wz;
  float n1 = cx * cx + cy * cy + cz * cz;
  float n2 = p2x * p2x + p2y * p2y + p2z * p2z;
  float nw = wx * wx + wy * wy + wz * wz;
  pc1T[i]  = make_float4(-2.f * cx,  -2.f * cy,  -2.f * cz,  n1);
  pc2T[i]  = make_float4(-2.f * p2x, -2.f * p2y, -2.f * p2z, n2);
  warpT[i] = make_float4(-2.f * wx,  -2.f * wy,  -2.f * wz,  nw);
  nwarp[i] = nw;
  npc2[i]  = n2;
}

// ---------------------------------------------------------------------------
// WMMA-tiled kNN: one wave owns 16 query rows. Per 16-point DB tile, one
// v_wmma_f32_16x16x4_f32 produces a 16x16 score tile; each lane holds one
// query column and 8 DB rows -> fully in-register branchless top-K, merged
// across the two half-wave streams once at the end.
// ---------------------------------------------------------------------------
template <int K, bool WRITE_D, bool WRITE_I>
__global__ __launch_bounds__(256) void knn_kernel(
    const float* __restrict__ qpts,   // (N,3) raw query coords
    const float* __restrict__ qn,     // (N) query squared norms (WRITE_D only)
    const float4* __restrict__ dbT,   // (N) prepared DB: (-2x,-2y,-2z,n2)
    int N, int* __restrict__ out_idx, float* __restrict__ out_dist) {
  __shared__ float4 sdb[256];
  __shared__ float smd[8][32][K];
  __shared__ int   smi[8][32][K];

  const int lane = threadIdx.x & 31;
  const int wave = threadIdx.x >> 5;
  const int hi   = lane >> 4;                 // 0: K0/K1 pair, 1: K2/K3 pair
  const int q    = (blockIdx.x * 8 + wave) * 16 + (lane & 15);
  const int qc   = (q < N) ? q : (N - 1);

  const float qx = qpts[3 * qc + 0];
  const float qy = qpts[3 * qc + 1];
  const float qz = qpts[3 * qc + 2];

  // B operand (4x16): rows = (qx,qy,qz,1) per query column.
  v2f bmat;
  bmat.x = hi ? qz : qx;
  bmat.y = hi ? 1.0f : qy;

  float kd[K];
  int   ki[K];
#pragma unroll
  for (int j = 0; j < K; ++j) { kd[j] = 3.0e38f; ki[j] = 0; }

  for (int cb = 0; cb < N; cb += 256) {
    __syncthreads();
    int gi = cb + threadIdx.x;
    sdb[threadIdx.x] = (gi < N) ? dbT[gi] : make_float4(0.f, 0.f, 0.f, 1.0e37f);
    __syncthreads();
    if (cb + 256 < N)
      __builtin_prefetch(&dbT[cb + 256 + threadIdx.x], 0, 1);  // global_prefetch_b8

    for (int t4 = 0; t4 < 16; t4 += 4) {
      // Batch 4 tiles of A-operands so the ds_load_b64s pipeline ahead of
      // the WMMAs (single dscnt wait instead of one per tile).
      v2f am[4];
#pragma unroll
      for (int u = 0; u < 4; ++u) {
        const float* sp = (const float*)&sdb[(t4 + u) * 16 + (lane & 15)];
        am[u].x = sp[hi * 2 + 0];
        am[u].y = sp[hi * 2 + 1];
      }
#pragma unroll
      for (int u = 0; u < 4; ++u) {
        float sc[8];
#ifdef USE_WMMA44
        v8f c = {};
        c = __builtin_amdgcn_wmma_f32_16x16x4_f32(
            false, am[u], false, bmat, (short)0, c, false, false);
#pragma unroll
        for (int j = 0; j < 8; ++j) sc[j] = c[j];
#else
#pragma unroll
        for (int j = 0; j < 8; ++j) {
          float4 p = sdb[(t4 + u) * 16 + hi * 8 + j];
          sc[j] = fmaf(p.x, qx, fmaf(p.y, qy, fmaf(p.z, qz, p.w)));
        }
#endif
        const int idbase = cb + (t4 + u) * 16 + hi * 8;
#pragma unroll
        for (int j = 0; j < 8; ++j) {
          float s  = sc[j];
          int   id = idbase + j;
          if (s < kd[K - 1]) {  // guarded, fully branchless bubble insert
            float d = s; int di = id;
#pragma unroll
            for (int p = 0; p < K; ++p) {
              bool  sw = d < kd[p];           // no tie-break: pure cmp+cndmask
              float td = sw ? kd[p] : d;
              int   ti = sw ? ki[p] : di;
              kd[p] = sw ? d : kd[p];
              ki[p] = sw ? di : ki[p];
              d = td; di = ti;
            }
          }
        }
      }
    }
  }

  // Merge the two half-wave candidate streams for each query.
  __syncthreads();
#pragma unroll
  for (int j = 0; j < K; ++j) { smd[wave][lane][j] = kd[j]; smi[wave][lane][j] = ki[j]; }
  __syncthreads();

  if (lane < 16 && q < N) {
    const float nq = WRITE_D ? qn[qc] : 0.0f;
    int a = 0, b = 0;
    for (int j = 0; j < K; ++j) {
      float da = smd[wave][lane][a], db = smd[wave][lane + 16][b];
      int   ia = smi[wave][lane][a], ib = smi[wave][lane + 16][b];
      bool  ta = da <= db;                    // branchless two-pointer merge
      float od = ta ? da : db;
      int   oi = ta ? ia : ib;
      if (ta) ++a; else ++b;
      if (WRITE_I) out_idx[q * K + j] = oi;
      if (WRITE_D) out_dist[q * K + j] = od + nq;
    }
  }
}

// ---------------------------------------------------------------------------
// curvature gather: (sum_{j<10} pts[idx_j] - 10*pts[i]) / 9
// ---------------------------------------------------------------------------
__global__ __launch_bounds__(256) void curv_kernel(
    const float* __restrict__ pts, const int* __restrict__ idx, int N,
    float* __restrict__ out) {
  int i = blockIdx.x * blockDim.x + threadIdx.x;
  if (i >= N) return;
  float sx = -10.f * pts[3 * i + 0];
  float sy = -10.f * pts[3 * i + 1];
  float sz = -10.f * pts[3 * i + 2];
#pragma unroll
  for (int j = 0; j < 10; ++j) {
    int id = idx[i * 10 + j];
    sx += pts[3 * id + 0];
    sy += pts[3 * id + 1];
    sz += pts[3 * id + 2];
  }
  const float inv9 = 1.0f / 9.0f;
  out[3 * i + 0] = sx * inv9;
  out[3 * i + 1] = sy * inv9;
  out[3 * i + 2] = sz * inv9;
}

// ---------------------------------------------------------------------------
// smoothness: sum_{j<k} ||flow[idx_j]-flow[i]|| / 8  (zero-safe for self)
// ---------------------------------------------------------------------------
__global__ __launch_bounds__(256) void smooth_kernel(
    const float* __restrict__ flow, const int* __restrict__ idx,
    const int* __restrict__ kptr, int N, float* __restrict__ out) {
  int i = blockIdx.x * blockDim.x + threadIdx.x;
  if (i >= N) return;
  int k = *kptr;
  if (k > 10) k = 10;
  float fx = flow[3 * i + 0], fy = flow[3 * i + 1], fz = flow[3 * i + 2];
  float s = 0.f;
  for (int j = 0; j < k; ++j) {
    int id = idx[i * 10 + j];
    float dx = flow[3 * id + 0] - fx;
    float dy = flow[3 * id + 1] - fy;
    float dz = flow[3 * id + 2] - fz;
    float sq = dx * dx + dy * dy + dz * dz;
    s += (sq > 0.f) ? sqrtf(sq) : 0.f;
  }
  out[i] = s * 0.125f;
}

// ---------------------------------------------------------------------------
// per-point loss terms + deterministic block tree reduction
// total_i = 0.02*(dist1+dist2) + 0.006*||interpCurv - movedCurv||^2 + 0.01*smooth
// ---------------------------------------------------------------------------
__global__ __launch_bounds__(256) void loss_kernel(
    const float* __restrict__ cdist, const int* __restrict__ cidx,
    const float* __restrict__ dist2, const float* __restrict__ curv2,
    const float* __restrict__ mcurv, const float* __restrict__ smoothv,
    int N, float* __restrict__ partial) {
  __shared__ float red[256];
  int i = blockIdx.x * 256 + threadIdx.x;
  float val = 0.f;
  if (i < N) {
    float d1 = cdist[i * 5 + 0];
    float w[5];
    float ws = 0.f;
#pragma unroll
    for (int j = 0; j < 5; ++j) { w[j] = 1.0f / (cdist[i * 5 + j] + 1e-8f); ws += w[j]; }
    float ix = 0.f, iy = 0.f, iz = 0.f;
#pragma unroll
    for (int j = 0; j < 5; ++j) {
      int id = cidx[i * 5 + j];
      float wj = w[j] / ws;
      ix += wj * curv2[3 * id + 0];
      iy += wj * curv2[3 * id + 1];
      iz += wj * curv2[3 * id + 2];
    }
    float dx = ix - mcurv[3 * i + 0];
    float dy = iy - mcurv[3 * i + 1];
    float dz = iz - mcurv[3 * i + 2];
    float cp = dx * dx + dy * dy + dz * dz;
    val = 0.02f * (d1 + dist2[i]) + 0.006f * cp + 0.01f * smoothv[i];
  }
  red[threadIdx.x] = val;
  __syncthreads();
  for (int s = 128; s > 0; s >>= 1) {
    if (threadIdx.x < s) red[threadIdx.x] += red[threadIdx.x + s];
    __syncthreads();
  }
  if (threadIdx.x == 0) partial[blockIdx.x] = red[0];
}

__global__ void final_kernel(const float* __restrict__ partial, int nb,
                             float* __restrict__ out) {
  if (threadIdx.x == 0) {
    float s = 0.f;
    for (int j = 0; j < nb; ++j) s += partial[j];  // fixed order: deterministic
    out[0] = s;
  }
}

// ---------------------------------------------------------------------------
extern "C" void kernel_launch(void* const* d_in, const int* in_sizes, int n_in,
                              void* d_out, int out_size, void* d_ws, size_t ws_size,
                              hipStream_t stream) {
  const float* pred   = (const float*)d_in[0];  // registration_pred (1,N,3)
  const float* gt     = (const float*)d_in[1];  // registration_gt   (1,N,3)
  const float* coords = (const float*)d_in[2];  // coords            (N,3)
  const int*   kptr   = (const int*)d_in[3];    // smoothness_k

  const int N = in_sizes[2] / 3;

  // workspace carve-up (float4 arrays first for 16B alignment)
  float4* pc1T  = (float4*)d_ws;             // N
  float4* pc2T  = pc1T + N;                  // N
  float4* warpT = pc2T + N;                  // N
  float*  pc2   = (float*)(warpT + N);       // 3N
  float*  warp  = pc2 + 3 * N;               // 3N
  float*  nwarp = warp + 3 * N;              // N
  float*  npc2  = nwarp + N;                 // N
  float*  curv2 = npc2 + N;                  // 3N
  float*  mcurv = curv2 + 3 * N;             // 3N
  float*  smoothv = mcurv + 3 * N;           // N
  float*  cdist = smoothv + N;               // 5N
  float*  dist2 = cdist + 5 * N;             // N
  int*    idx1  = (int*)(dist2 + N);         // 10N
  int*    idx2  = idx1 + 10 * N;             // 10N
  int*    cidx  = idx2 + 10 * N;             // 5N
  float*  partial = (float*)(cidx + 5 * N);  // nb

  const int nb   = (N + 255) / 256;
  const int knng = (N + 127) / 128;          // 8 waves/block * 16 queries/wave

  prep_kernel<<<nb, 256, 0, stream>>>(pred, gt, coords, N, pc2, warp,
                                      pc1T, pc2T, warpT, nwarp, npc2);

  // self-kNN pc1 (k=10): indices only
  knn_kernel<10, false, true><<<knng, 256, 0, stream>>>(
      coords, nullptr, pc1T, N, idx1, nullptr);
  // self-kNN pc2 (k=10): indices only
  knn_kernel<10, false, true><<<knng, 256, 0, stream>>>(
      pc2, nullptr, pc2T, N, idx2, nullptr);
  // cross warp -> pc2 (k=5): dists+indices (dist1 = cdist[i*5+0])
  knn_kernel<5, true, true><<<knng, 256, 0, stream>>>(
      warp, nwarp, pc2T, N, cidx, cdist);
  // cross pc2 -> warp (k=1): min distance only (dist2)
  knn_kernel<1, true, false><<<knng, 256, 0, stream>>>(
      pc2, npc2, warpT, N, nullptr, dist2);

  curv_kernel<<<nb, 256, 0, stream>>>(pc2, idx2, N, curv2);    // pc2 curvature
  curv_kernel<<<nb, 256, 0, stream>>>(warp, idx1, N, mcurv);   // moved curvature
  smooth_kernel<<<nb, 256, 0, stream>>>(pred, idx1, kptr, N, smoothv);

  loss_kernel<<<nb, 256, 0, stream>>>(cdist, cidx, dist2, curv2, mcurv,
                                      smoothv, N, partial);
  final_kernel<<<1, 32, 0, stream>>>(partial, nb, (float*)d_out);
}